// Model_1666447311107
// MI455X (gfx1250) — compile-verified
//
#include <hip/hip_runtime.h>

// Problem constants (match reference)
#define N_USER   200000
#define N_ITEM   300000
#define LATDIM   64
#define N_NODES  (N_USER + N_ITEM)
#define N_EDGES  16000000
#define N_LAYERS 3

// Flat float counts
#define NODE_FLOATS ((long)N_NODES * LATDIM)   // 32,000,000 floats = 128 MB
#define NODE_QUADS  (NODE_FLOATS / 4)          // 8,000,000 float4

// SpMM tiling: wave-private async staging of the edge stream
#define EDGES_PER_CHUNK 128                    // 512 B per array per chunk
#define CHUNKS_PER_WAVE 5
#define EDGES_PER_WAVE  (EDGES_PER_CHUNK * CHUNKS_PER_WAVE)   // 640
#define WAVES_PER_BLOCK 8
#define SPMM_BLOCKS     (N_EDGES / (EDGES_PER_WAVE * WAVES_PER_BLOCK)) // 3125 exact

// ---------------------------------------------------------------------------
// LDS byte offset of a generic pointer to a __shared__ object:
// addrspacecast flat->local (AS3), then ptrtoint (AS3 pointers are 32-bit).
// ---------------------------------------------------------------------------
__device__ __forceinline__ unsigned lds_addr_of(void* p) {
    return (unsigned)(unsigned long long)(__attribute__((address_space(3))) char*)p;
}

// ---------------------------------------------------------------------------
// Async global->LDS copy, 16 B per lane (gfx1250 GLOBAL_LOAD_ASYNC_TO_LDS_B128,
// tracked by ASYNCcnt). NT hint: the edge stream is read exactly once per
// layer (576 MB total) and must not evict the L2-resident x/y matrices.
// ---------------------------------------------------------------------------
__device__ __forceinline__ void async_copy_b128(void* lds_dst, const void* gsrc) {
    unsigned l = lds_addr_of(lds_dst);
    asm volatile("global_load_async_to_lds_b128 %0, %1, off th:TH_LOAD_NT"
                 :: "v"(l), "v"(gsrc) : "memory");
}

__device__ __forceinline__ void wait_async_le3() {
    asm volatile("s_wait_asynccnt 0x3" ::: "memory");
}
__device__ __forceinline__ void wait_async_0() {
    asm volatile("s_wait_asynccnt 0x0" ::: "memory");
}

// ---------------------------------------------------------------------------
// No-return fp32 global atomic add (HW instruction, STOREcnt path, never a
// CAS loop). Syntax verified by the round-1 compile. S_ENDPGM's implicit
// wait-idle covers completion; VMEM atomics snapshot VGPRs at issue.
// ---------------------------------------------------------------------------
__device__ __forceinline__ void atomic_add_f32_nr(float* p, float v) {
    asm volatile("global_atomic_add_f32 %0, %1, off" :: "v"(p), "v"(v) : "memory");
}

// ---------------------------------------------------------------------------
// acc = concat(u, i); xbuf = concat(u, i)     (float4-vectorized)
// ---------------------------------------------------------------------------
__global__ __launch_bounds__(256) void init_concat(
    const float4* __restrict__ u, const float4* __restrict__ it,
    float4* __restrict__ acc, float4* __restrict__ xbuf)
{
    long i = (long)blockIdx.x * blockDim.x + threadIdx.x;
    if (i >= NODE_QUADS) return;
    const long uq = (long)N_USER * (LATDIM / 4);
    float4 v = (i < uq) ? u[i] : it[i - uq];
    acc[i]  = v;
    xbuf[i] = v;
}

// ---------------------------------------------------------------------------
// y = 0
// ---------------------------------------------------------------------------
__global__ __launch_bounds__(256) void zero_quads(float4* __restrict__ y)
{
    long i = (long)blockIdx.x * blockDim.x + threadIdx.x;
    if (i >= NODE_QUADS) return;
    y[i] = make_float4(0.f, 0.f, 0.f, 0.f);
}

// ---------------------------------------------------------------------------
// COO SpMM: y[rows[e]] += vals[e] * x[cols[e]]
//
// Each wave owns 640 consecutive edges. The edge stream (rows/cols/vals) is
// staged into wave-private LDS with double-buffered async B128 copies
// (3 instructions per 128-edge chunk), ordered solely by s_wait_asynccnt
// (only the producing wave reads its slice -> no barriers).
// Processing: one edge per 16-lane half-group; lane l gathers float4 #l of
// x[col] (256 B coalesced) and issues 4 no-return global_atomic_add_f32.
// ---------------------------------------------------------------------------
__global__ __launch_bounds__(256) void spmm_edges_async(
    const int*   __restrict__ rows,
    const int*   __restrict__ cols,
    const float* __restrict__ vals,
    const float* __restrict__ x,
    float*       __restrict__ y)
{
    // [wave][buffer][array: 0=rows 1=cols 2=vals][edge-in-chunk]
    __shared__ int smem[WAVES_PER_BLOCK][2][3][EDGES_PER_CHUNK];

    const unsigned lane = threadIdx.x & 31u;
    const unsigned wv   = threadIdx.x >> 5;
    const unsigned base = (blockIdx.x * WAVES_PER_BLOCK + wv) * EDGES_PER_WAVE;

    const unsigned l16 = lane & 15u;   // float4 slot within the 64-float row
    const unsigned h   = lane >> 4;    // which of the 2 edges this half-wave owns

    // Prologue: stage chunk 0 into buffer 0
    {
        const unsigned g = base + lane * 4u;      // 4 ints (16 B) per lane
        async_copy_b128(&smem[wv][0][0][lane * 4u], rows + g);
        async_copy_b128(&smem[wv][0][1][lane * 4u], cols + g);
        async_copy_b128(&smem[wv][0][2][lane * 4u], vals + g);
    }

    for (int c = 0; c < CHUNKS_PER_WAVE; ++c) {
        const int buf = c & 1;
        if (c + 1 < CHUNKS_PER_WAVE) {
            // Stage next chunk into the other buffer, then wait for current.
            const unsigned g = base + (unsigned)(c + 1) * EDGES_PER_CHUNK + lane * 4u;
            async_copy_b128(&smem[wv][buf ^ 1][0][lane * 4u], rows + g);
            async_copy_b128(&smem[wv][buf ^ 1][1][lane * 4u], cols + g);
            async_copy_b128(&smem[wv][buf ^ 1][2][lane * 4u], vals + g);
            wait_async_le3();   // the 3 newest remain in flight; chunk c is done
        } else {
            wait_async_0();
        }

        #pragma unroll 2
        for (int s = 0; s < EDGES_PER_CHUNK / 2; ++s) {
            const unsigned slot = 2u * (unsigned)s + h;
            const int   r = smem[wv][buf][0][slot];
            const int   cl = smem[wv][buf][1][slot];
            const float w = __int_as_float(smem[wv][buf][2][slot]);

            const float4* xr = (const float4*)(x + (size_t)cl * LATDIM);
            const float4  xv = xr[l16];                 // 256 B coalesced gather

            float* yr = y + (size_t)r * LATDIM + (size_t)l16 * 4u;
            atomic_add_f32_nr(yr + 0, w * xv.x);
            atomic_add_f32_nr(yr + 1, w * xv.y);
            atomic_add_f32_nr(yr + 2, w * xv.z);
            atomic_add_f32_nr(yr + 3, w * xv.w);
        }
    }
}

// ---------------------------------------------------------------------------
// acc += y
// ---------------------------------------------------------------------------
__global__ __launch_bounds__(256) void acc_add(
    float4* __restrict__ acc, const float4* __restrict__ y)
{
    long i = (long)blockIdx.x * blockDim.x + threadIdx.x;
    if (i >= NODE_QUADS) return;
    float4 a = acc[i];
    float4 b = y[i];
    a.x += b.x; a.y += b.y; a.z += b.z; a.w += b.w;
    acc[i] = a;
}

// ---------------------------------------------------------------------------
// Host-side orchestration (graph-capture safe: only kernel launches on stream)
// ---------------------------------------------------------------------------
extern "C" void kernel_launch(void* const* d_in, const int* in_sizes, int n_in,
                              void* d_out, int out_size, void* d_ws, size_t ws_size,
                              hipStream_t stream)
{
    const int*   rows = (const int*)  d_in[0];
    const int*   cols = (const int*)  d_in[1];
    const float* vals = (const float*)d_in[2];
    const float* uE   = (const float*)d_in[3];
    const float* iE   = (const float*)d_in[4];

    float* acc = (float*)d_out;                                       // [N, 64]
    float* x   = (float*)d_ws;                                        // 128 MB
    float* y   = (float*)((char*)d_ws + NODE_FLOATS * sizeof(float)); // 128 MB

    const unsigned nodeBlocks = (unsigned)((NODE_QUADS + 255) / 256);

    init_concat<<<nodeBlocks, 256, 0, stream>>>(
        (const float4*)uE, (const float4*)iE, (float4*)acc, (float4*)x);

    for (int layer = 0; layer < N_LAYERS; ++layer) {
        zero_quads<<<nodeBlocks, 256, 0, stream>>>((float4*)y);
        spmm_edges_async<<<SPMM_BLOCKS, 256, 0, stream>>>(rows, cols, vals, x, y);
        acc_add<<<nodeBlocks, 256, 0, stream>>>((float4*)acc, (const float4*)y);
        float* t = x; x = y; y = t;   // next layer's input is this layer's output
    }
}